// XCorrExt_34205119545375
// MI455X (gfx1250) — compile-verified
//
#include <hip/hip_runtime.h>

typedef __attribute__((ext_vector_type(2))) float v2f;
typedef __attribute__((ext_vector_type(8))) float v8f;

#define S_LEN   160000
#define FN      1000
#define N_WIN   320
#define HOP     160
#define TAU     257
#define LAG_CUT 33
#define OUT_LAGS 224      // TAU - LAG_CUT
#define EXT_LEN 576       // N + TAU - 1
#define EXT_PAD 608       // zero-padded so A reads up to index 590 are safe
#define K_GROUPS 35       // K = 560 = 16 * 35 (4 WMMA of K=4 per group)
#define AZ_BIAS 240       // aZ[i] = a[i - 240], i in [0,800)
#define AZ_SWZ  852       // swizzled size: swz(799)=848 used (+pad)
#define WAVE_LDS 2080     // ext(608)+aZ(852)+cs(580)+sums(32)=2072 -> 2080

// One wave (32 lanes) computes one frame:
//   C[j,q] = sum_k ext[16+k+j] * a[k-16q]  ==>  num[tau], tau = 16 + 16q + j
// via 140 x V_WMMA_F32_16X16X4_F32 (exact f32, matches the f32 reference).
// Operands come from zero-padded LDS arrays (no predication); B uses a
// +1-per-16 bank swizzle (conflict-free for the -16 lane stride) that keeps
// operand pairs adjacent, so every WMMA needs exactly 2x ds_load_2addr_b32.
__global__ __launch_bounds__(128)
void XCorrExt_34205119545375_kernel(const float* __restrict__ x,
                                    float* __restrict__ out) {
    __shared__ float smem[4 * WAVE_LDS];
    const int lane = threadIdx.x & 31;
    const int wave = threadIdx.x >> 5;
    const int g = blockIdx.x * 4 + wave;     // global frame id, < 32000
    const int b = g / FN;
    const int f = g - b * FN;

    float* ext  = smem + wave * WAVE_LDS;    // ext window, 608 floats (>=576 zeroed)
    float* aZ   = ext + EXT_PAD;             // zero-padded swizzled frame, 852 floats
    float* cs   = aZ + AZ_SWZ;               // cumsum of ext^2, 577 floats (pad 580)
    float* sums = cs + 580;                  // 32 chunk totals

    const float* xb = x + (size_t)b * S_LEN;

    // ---- stage 1a: ext[0..575] = frame f ++ first 256 of frame f+1 (wrap), pad 0
    for (int j = lane; j < EXT_PAD; j += 32) {
        float v = 0.0f;
        if (j < N_WIN) {
            int idx = f * HOP + j;                       // xp zero-pads past S
            if (idx < S_LEN) v = xb[idx];
        } else if (j < EXT_LEN) {
            int fn  = (f + 1 == FN) ? 0 : f + 1;         // wrap of xf
            int idx = fn * HOP + (j - N_WIN);
            if (idx < S_LEN) v = xb[idx];
        }
        ext[j] = v;
    }
    // ---- stage 1b: aZ[swz(i)] = a[i-240] (zeros outside), swz(i) = i + (i>>4).
    // Pad slots (off % 17 == 16) are never read, so no div/mod and no pad fill.
    for (int i = lane; i < 800; i += 32) {
        float v = 0.0f;
        int n = i - AZ_BIAS;
        if (n >= 0 && n < N_WIN) {
            int idx = f * HOP + n;
            if (idx < S_LEN) v = xb[idx];
        }
        aZ[i + (i >> 4)] = v;
    }
    __syncthreads();

    // ---- stage 2: cs[i] = sum_{t<i} ext[t]^2, i in [0,576]  (32 chunks of 18)
    float loc[18];
    {
        const int j0 = lane * 18;
        float acc = 0.0f;
        #pragma unroll
        for (int t = 0; t < 18; ++t) {
            float v = ext[j0 + t];
            acc += v * v;
            loc[t] = acc;
        }
        sums[lane] = acc;
    }
    __syncthreads();
    {
        float pre = 0.0f;
        for (int l = 0; l < 32; ++l) {          // broadcast LDS reads
            float t = sums[l];
            if (l < lane) pre += t;
        }
        if (lane == 0) cs[0] = 0.0f;
        const int j0 = lane * 18;
        #pragma unroll
        for (int t = 0; t < 18; ++t) cs[j0 + 1 + t] = pre + loc[t];
    }
    __syncthreads();

    // ---- stage 3: Toeplitz GEMM on the f32 matrix pipe
    const int r2 = lane >> 4;                 // lane-half selects K, K+2 (ISA 16x4 A layout)
    const int lo = lane & 15;
    const int ib = AZ_BIAS + 2 * r2 - 16 * lo;        // i_base in [0,242], (ib&15) in {0,2}
    const float* pA = ext + 16 + lo + 2 * r2;         // A[j,k] = ext[16 + k + j]
    const float* pB = aZ + ib + (ib >> 4);            // swizzled, advances +17/group

    v8f c = {0.f, 0.f, 0.f, 0.f, 0.f, 0.f, 0.f, 0.f};
    #pragma unroll 5
    for (int t = 0; t < K_GROUPS; ++t) {
        v2f a, bb;
        a.x = pA[0];  a.y = pA[1];  bb.x = pB[0];  bb.y = pB[1];
        c = __builtin_amdgcn_wmma_f32_16x16x4_f32(false, a, false, bb, (short)0, c, false, false);
        a.x = pA[4];  a.y = pA[5];  bb.x = pB[4];  bb.y = pB[5];
        c = __builtin_amdgcn_wmma_f32_16x16x4_f32(false, a, false, bb, (short)0, c, false, false);
        a.x = pA[8];  a.y = pA[9];  bb.x = pB[8];  bb.y = pB[9];
        c = __builtin_amdgcn_wmma_f32_16x16x4_f32(false, a, false, bb, (short)0, c, false, false);
        a.x = pA[12]; a.y = pA[13]; bb.x = pB[12]; bb.y = pB[13];
        c = __builtin_amdgcn_wmma_f32_16x16x4_f32(false, a, false, bb, (short)0, c, false, false);
        pA += 16;                 // k0 advances 16 per group
        pB += 17;                 // swizzled offset advances 17 per group
    }

    // ---- stage 4: normalize + store. C layout: VGPR r -> M = r + 8*r2, N = lo.
    const float e0 = cs[N_WIN];
    float* ob = out + ((size_t)b * FN + f) * OUT_LAGS;
    #pragma unroll
    for (int r = 0; r < 8; ++r) {
        const int j   = r + 8 * r2;
        const int tau = 16 + 16 * lo + j;     // distinct taus 16..271 across the tile
        if (tau >= LAG_CUT && tau < TAU) {
            const float et  = cs[tau + N_WIN] - cs[tau];
            const float den = e0 + et + 1e-5f;
            ob[tau - LAG_CUT] = 2.0f * c[r] * __builtin_amdgcn_rcpf(den);
        }
    }
}

extern "C" void kernel_launch(void* const* d_in, const int* in_sizes, int n_in,
                              void* d_out, int out_size, void* d_ws, size_t ws_size,
                              hipStream_t stream) {
    const float* x = (const float*)d_in[0];
    float* out = (float*)d_out;
    // 32 batches * 1000 frames = 32000 frames; 4 waves (frames) per 128-thread block
    dim3 grid(8000), block(128);
    hipLaunchKernelGGL(XCorrExt_34205119545375_kernel, grid, block, 0, stream, x, out);
}